// USMSharp_49667001811110
// MI455X (gfx1250) — compile-verified
//
#include <hip/hip_runtime.h>

// USM sharpen: sharp = 1.5*img - 0.5*(img (*) G51x51), reflect padding.
// G = outer(g,g) -> separable: two 51-tap 1D passes, each a banded matmul
// done with chained V_WMMA_F32_16X16X4_F32 (K padded 66 -> 68 = 17 chunks).
//
// Layouts (ISA 7.12.2, wave32):
//   A 16x4 f32 : lane(0-15)=row M, v0/v1 = K{0,1}; lanes 16-31 = K{2,3}
//   B 4x16 f32 : lane(0-15)=col N, v0/v1 = K{0,1}; lanes 16-31 = K{2,3}
//   C/D 16x16  : VGPR v -> M=v (lanes 0-15) / M=v+8 (lanes 16-31), N=lane&15

typedef __attribute__((ext_vector_type(2))) float v2f;
typedef __attribute__((ext_vector_type(8))) float v8f;

#define KRAD   25
#define KLEN   51
#define NCHUNK 17     // ceil(66/4); G rows >= 66 are zero via padded table
#define GOFF   16     // s_gp[GOFF + i] = g[i] for i in [0,51), 0 elsewhere
#define GPAD   96     // indices used: GOFF + (j0 - m) in [1, 84]
#define WDIM   512
#define HDIM   512

__device__ __forceinline__ int reflect_i(int i, int n) {
    // jnp.pad "reflect" (no edge repeat); p=25 < n so one reflection suffices
    i = (i < 0) ? -i : i;
    i = (i >= n) ? (2 * n - 2 - i) : i;
    return i;
}

// Recover 1D Gaussian from the 2D kernel via row sums (g sums to 1),
// stored zero-padded so banded-matrix gathers need no bounds checks.
__device__ __forceinline__ void load_gpad(const float* __restrict__ k2d, float* s_gp) {
    for (int idx = threadIdx.x; idx < GPAD; idx += blockDim.x) {
        const int i = idx - GOFF;
        float s = 0.f;
        if ((unsigned)i < (unsigned)KLEN) {
            for (int j = 0; j < KLEN; ++j) s += k2d[i * KLEN + j];
        }
        s_gp[idx] = s;
    }
    __syncthreads();
}

// ---------------- Pass 1: horizontal 51-tap blur -> tmp ----------------
// D(16x16) = A(16x68) * G(68x16); A[m][j] = in[y0+m][x0-25+j] (x reflect),
// G[j][n] = g[j-n].
__global__ __launch_bounds__(256) void usm_hblur(
    const float* __restrict__ img, const float* __restrict__ k2d,
    float* __restrict__ tmp, int nplanes)
{
    __shared__ float s_gp[GPAD];
    load_gpad(k2d, s_gp);

    const int lane = threadIdx.x & 31;
    const int wv   = threadIdx.x >> 5;
    const int tilesPerPlane = (WDIM >> 4) * (HDIM >> 4);   // 32*32 = 1024
    const int tile = blockIdx.x * (blockDim.x >> 5) + wv;
    if (tile >= nplanes * tilesPerPlane) return;

    const int p  = tile >> 10;
    const int t  = tile & 1023;
    const int y0 = (t >> 5) << 4;
    const int x0 = (t & 31) << 4;

    const int m  = lane & 15;       // A row M / B col N
    const int hi = lane >> 4;       // K sub-pair select: {0,1} vs {2,3}
    const float* inr = img + p * (HDIM * WDIM) + (y0 + m) * WDIM;
    const float* gb  = s_gp + (GOFF + 2 * hi - m);   // gb[4k] = g[4k+2hi-m]

    v8f acc = {};
    if (x0 >= KRAD && x0 + 42 < WDIM) {
        // Interior tile (wave-uniform): no reflection, immediate-offset loads
        const float* ap = inr + (x0 - KRAD + 2 * hi);
        #pragma unroll
        for (int k = 0; k < NCHUNK; ++k) {
            v2f a; a.x = ap[4 * k];  a.y = ap[4 * k + 1];
            v2f b; b.x = gb[4 * k];  b.y = gb[4 * k + 1];
            acc = __builtin_amdgcn_wmma_f32_16x16x4_f32(
                false, a, false, b, (short)0, acc, false, false);
        }
    } else {
        #pragma unroll
        for (int k = 0; k < NCHUNK; ++k) {
            const int c0 = x0 - KRAD + 4 * k + 2 * hi;
            v2f a;
            a.x = inr[reflect_i(c0,     WDIM)];
            a.y = inr[reflect_i(c0 + 1, WDIM)];
            v2f b; b.x = gb[4 * k];  b.y = gb[4 * k + 1];
            acc = __builtin_amdgcn_wmma_f32_16x16x4_f32(
                false, a, false, b, (short)0, acc, false, false);
        }
    }

    float* op = tmp + p * (HDIM * WDIM) + (y0 + 8 * hi) * WDIM + x0 + m;
    #pragma unroll
    for (int v = 0; v < 8; ++v)
        op[v * WDIM] = acc[v];
}

// ------------- Pass 2: vertical 51-tap blur + fused sharpen -------------
// D(16x16) = Gv(16x68) * T(68x16); Gv[m][j] = g[j-m],
// T[j][n] = tmp[reflect(y0-25+j)][x0+n].  out = 1.5*img - 0.5*D.
__global__ __launch_bounds__(256) void usm_vblur_sharp(
    const float* __restrict__ tmp, const float* __restrict__ img,
    const float* __restrict__ k2d, float* __restrict__ out, int nplanes)
{
    __shared__ float s_gp[GPAD];
    load_gpad(k2d, s_gp);

    const int lane = threadIdx.x & 31;
    const int wv   = threadIdx.x >> 5;
    const int tilesPerPlane = (WDIM >> 4) * (HDIM >> 4);
    const int tile = blockIdx.x * (blockDim.x >> 5) + wv;
    if (tile >= nplanes * tilesPerPlane) return;

    const int p  = tile >> 10;
    const int t  = tile & 1023;
    const int y0 = (t >> 5) << 4;
    const int x0 = (t & 31) << 4;

    const int m  = lane & 15;
    const int hi = lane >> 4;
    const float* tp = tmp + p * (HDIM * WDIM);
    const float* gb = s_gp + (GOFF + 2 * hi - m);    // Gv[m][4k+2hi]

    v8f acc = {};
    if (y0 >= KRAD && y0 + 42 < HDIM) {
        // Interior tile: B loads stride by W with immediate offsets
        const float* bp = tp + (y0 - KRAD + 2 * hi) * WDIM + x0 + m;
        #pragma unroll
        for (int k = 0; k < NCHUNK; ++k) {
            v2f a; a.x = gb[4 * k];             a.y = gb[4 * k + 1];
            v2f b; b.x = bp[(4 * k) * WDIM];    b.y = bp[(4 * k + 1) * WDIM];
            acc = __builtin_amdgcn_wmma_f32_16x16x4_f32(
                false, a, false, b, (short)0, acc, false, false);
        }
    } else {
        #pragma unroll
        for (int k = 0; k < NCHUNK; ++k) {
            const int j0 = 4 * k + 2 * hi;
            const int r0 = reflect_i(y0 - KRAD + j0,     HDIM);
            const int r1 = reflect_i(y0 - KRAD + j0 + 1, HDIM);
            v2f a; a.x = gb[4 * k];              a.y = gb[4 * k + 1];
            v2f b; b.x = tp[r0 * WDIM + x0 + m]; b.y = tp[r1 * WDIM + x0 + m];
            acc = __builtin_amdgcn_wmma_f32_16x16x4_f32(
                false, a, false, b, (short)0, acc, false, false);
        }
    }

    const int eoff = p * (HDIM * WDIM) + (y0 + 8 * hi) * WDIM + x0 + m;
    const float* ip = img + eoff;
    float*       op = out + eoff;
    #pragma unroll
    for (int v = 0; v < 8; ++v)
        op[v * WDIM] = 1.5f * ip[v * WDIM] - 0.5f * acc[v];
}

extern "C" void kernel_launch(void* const* d_in, const int* in_sizes, int n_in,
                              void* d_out, int out_size, void* d_ws, size_t ws_size,
                              hipStream_t stream) {
    const float* img = (const float*)d_in[0];   // (8,3,512,512) fp32
    const float* k2d = (const float*)d_in[1];   // (51,51) fp32
    float* out = (float*)d_out;
    float* tmp = (float*)d_ws;                  // 24*512*512 fp32 = 25.2 MB

    const int nplanes = in_sizes[0] / (HDIM * WDIM);   // 8*3 = 24
    const int tiles = nplanes * (HDIM / 16) * (WDIM / 16);
    const int wavesPerBlock = 8;                        // 256 threads = 8 wave32
    const int blocks = (tiles + wavesPerBlock - 1) / wavesPerBlock;

    usm_hblur<<<blocks, 256, 0, stream>>>(img, k2d, tmp, nplanes);
    usm_vblur_sharp<<<blocks, 256, 0, stream>>>(tmp, img, k2d, out, nplanes);
}